// MOEGate_42786464203493
// MI455X (gfx1250) — compile-verified
//
#include <hip/hip_runtime.h>
#include <hip/hip_bf16.h>
#include <math.h>
#include <stdint.h>

// ---------------------------------------------------------------------------
// MoE gate: logits = r @ W^T + b ; softmax ; top-8 mask ; renormalize.
//   r: (32768, 2048) f32, W: (64, 2048) f32, b: (64) f32
//   out: [weights_hard (32768x64) | weights_soft (32768x64)] f32
// GEMM: V_WMMA_F32_16X16X4_F32 (exact fp32). W staged to LDS by the Tensor
// Data Mover (TENSOR_LOAD_TO_LDS, double-buffered, HW padding for bank-
// conflict-free fragment reads). A fragments software-pipelined 4 deep.
// ---------------------------------------------------------------------------

typedef float        v2f __attribute__((ext_vector_type(2)));
typedef float        v8f __attribute__((ext_vector_type(8)));
typedef unsigned int v4u __attribute__((ext_vector_type(4)));
typedef int          v4i __attribute__((ext_vector_type(4)));
typedef int          v8i __attribute__((ext_vector_type(8)));

#define B_ROWS        32768
#define DDIM          2048
#define NEXP          64
#define TOPK          8
#define KC            256                   // K-chunk staged in LDS
#define NCHUNK        (DDIM / KC)           // 8
#define LDS_STRIDE    (KC + 4)              // 260: TDM pads 4 dwords / 256 dwords
#define BUF_FLOATS    (NEXP * LDS_STRIDE)   // 16640 floats = 65 KB + pad
#define BUF_BYTES     (BUF_FLOATS * 4)
#define ROWS_PER_BLK  128                   // 8 waves x 16 rows
#define THREADS       256

#if __has_builtin(__builtin_amdgcn_tensor_load_to_lds)
#define USE_TDM 1
#else
#define USE_TDM 0
#endif

__global__ __launch_bounds__(THREADS, 2)
void moe_gate_wmma_kernel(const float* __restrict__ r,
                          const float* __restrict__ W,
                          const float* __restrict__ bias,
                          float* __restrict__ out_hard,
                          float* __restrict__ out_soft)
{
    extern __shared__ __align__(16) float smem[];   // 2 x 66560 B (double buffer)

    const int tid  = threadIdx.x;
    const int lane = tid & 31;          // wave32
    const int wv   = tid >> 5;          // wave id 0..7
    const int half = lane >> 4;         // selects K pair inside fragment
    const int l16  = lane & 15;

    const int rowBlock = blockIdx.x * ROWS_PER_BLK;
    const int row0     = rowBlock + wv * 16;

    // A-fragment pointer: row (row0+l16), K offset 2*half.
    // ISA 16x4 f32 A layout: VGPR0 = K{0|2}, VGPR1 = K{1|3} by lane half.
    const float* aPtr = r + (size_t)(row0 + l16) * DDIM + 2 * half;

    // Per-lane bias for the 4 N-tiles (expert = t*16 + l16)
    float bReg[4];
    #pragma unroll
    for (int t = 0; t < 4; ++t) bReg[t] = bias[t * 16 + l16];

    v8f acc[4];
    #pragma unroll
    for (int t = 0; t < 4; ++t)
        #pragma unroll
        for (int i = 0; i < 8; ++i) acc[t][i] = 0.0f;

    // 4 WMMAs (one per 16-expert N-tile) against one A fragment.
    auto wmma4 = [&acc](v2f a, const float* bB, int kk) {
        #pragma unroll
        for (int t = 0; t < 4; ++t) {
            v2f bf = *(const v2f*)(bB + t * 16 * LDS_STRIDE + kk);
            acc[t] = __builtin_amdgcn_wmma_f32_16x16x4_f32(
                /*neg_a=*/false, a, /*neg_b=*/false, bf,
                /*c_mod=*/(short)0, acc[t],
                /*reuse_a=*/false, /*reuse_b=*/false);
        }
    };

#if USE_TDM
    // ---- Tensor Data Mover: one descriptor moves W[0:64][c*KC : (c+1)*KC)
    //      global->LDS with 4-dword pad every 256 dwords (row stride 260).
    const uint32_t ldsBase = (uint32_t)(uintptr_t)(const void*)smem; // LDS aperture: low 32 bits = LDS offset
    auto tdm_issue = [&](int c, int buf) {
        const uint64_t gaddr = (uint64_t)(uintptr_t)W + (uint64_t)c * (KC * 4);
        const uint32_t laddr = ldsBase + (uint32_t)buf * BUF_BYTES;
        // D# group 0: count=1 | lds_addr | global_addr[56:0] | type=2
        v4u g0 = { 1u, laddr, (uint32_t)gaddr,
                   (uint32_t)(gaddr >> 32) | (2u << 30) };
        // D# group 1: mask=0, data_size=4B, pad_en, pad_interval=256dw,
        //             pad_amount=4dw | dim0=256 | dim1=64 | tile 256x64 | stride 2048
        v8i g1 = { (int)((2u << 16) | (1u << 20) | (7u << 22) | (3u << 25)),
                   (int)(KC << 16),        // tensor_dim0[15:0] << 16
                   (int)(NEXP << 16),      // tensor_dim1[15:0] << 16
                   (int)(KC << 16),        // tile_dim0 << 16
                   NEXP,                   // tile_dim1 (tile_dim2 = 0)
                   DDIM,                   // tensor_dim0_stride[31:0]
                   0, 0 };
        v4i g2 = { 0, 0, 0, 0 };           // 2D: dims 2/3 unused
        v4i g3 = { 0, 0, 0, 0 };
#if defined(__clang_major__) && (__clang_major__ >= 23)
        v8i g4 = { 0, 0, 0, 0, 0, 0, 0, 0 };
        __builtin_amdgcn_tensor_load_to_lds(g0, g1, g2, g3, g4, 0);
#else
        __builtin_amdgcn_tensor_load_to_lds(g0, g1, g2, g3, 0);
#endif
    };

    if (wv == 0) tdm_issue(0, 0);               // fill buffer 0
    __builtin_amdgcn_s_wait_tensorcnt((short)0);
    __syncthreads();
#else
    // Fallback: cooperative VMEM->VGPR->LDS staging (no TDM builtin).
    auto coop_load = [&](int c, int buf) {
        float* dst = smem + buf * BUF_FLOATS;
        #pragma unroll
        for (int i = 0; i < (NEXP * KC / 4) / THREADS; ++i) {   // 16 float4/thread
            int idx = tid + i * THREADS;
            int e   = idx >> 6;               // KC/4 = 64 float4 per expert row
            int k4  = idx & 63;
            float4 v = *(const float4*)(W + (size_t)e * DDIM + c * KC + k4 * 4);
            *(float4*)(dst + e * LDS_STRIDE + k4 * 4) = v;
        }
    };
#endif

    // ----------------------------- GEMM main loop ---------------------------
    for (int c = 0; c < NCHUNK; ++c) {
#if USE_TDM
        if (wv == 0 && c + 1 < NCHUNK)
            tdm_issue(c + 1, (c + 1) & 1);      // async prefetch next W chunk
#else
        coop_load(c, c & 1);
        __syncthreads();
#endif
        const float* bB     = smem + (c & 1) * BUF_FLOATS + l16 * LDS_STRIDE + 2 * half;
        const float* aChunk = aPtr + c * KC;

        // prefetch this lane's A rows for the next chunk (WGP scope)
        if (c + 1 < NCHUNK)
            __builtin_prefetch(aChunk + KC, 0, 3);

        // software-pipelined A fragments, depth 4 (16 WMMAs of slack/load)
        v2f a0 = *(const v2f*)(aChunk + 0);
        v2f a1 = *(const v2f*)(aChunk + 4);
        v2f a2 = *(const v2f*)(aChunk + 8);
        v2f a3 = *(const v2f*)(aChunk + 12);
        #pragma unroll 4
        for (int kk = 0; kk < KC - 16; kk += 4) {
            v2f an = *(const v2f*)(aChunk + kk + 16);
            wmma4(a0, bB, kk);
            a0 = a1; a1 = a2; a2 = a3; a3 = an;
        }
        wmma4(a0, bB, KC - 16);
        wmma4(a1, bB, KC - 12);
        wmma4(a2, bB, KC - 8);
        wmma4(a3, bB, KC - 4);

#if USE_TDM
        __builtin_amdgcn_s_wait_tensorcnt((short)0);  // next buffer filled
#endif
        __syncthreads();                              // + this buffer free
    }

    // ----------------- Spill logits (+bias) to LDS (reuse buffer 0) ----------
    // C/D layout: lane<16 -> rows 0..7 (VGPR v = row v), lane>=16 -> rows 8..15
    #pragma unroll
    for (int t = 0; t < 4; ++t)
        #pragma unroll
        for (int v = 0; v < 8; ++v) {
            int rloc = wv * 16 + v + 8 * half;
            int e    = t * 16 + l16;
            smem[rloc * NEXP + e] = acc[t][v] + bReg[t];
        }
    __syncthreads();

    // ----------------- Softmax + top-8 + renormalize (1 thread / row) --------
    if (tid < ROWS_PER_BLK) {
        const float* lrow = &smem[tid * NEXP];
        const int    grow = rowBlock + tid;

        float m = -INFINITY;
        #pragma unroll 8
        for (int e = 0; e < NEXP; ++e) m = fmaxf(m, lrow[e]);

        float sum = 0.0f;
        #pragma unroll 8
        for (int e = 0; e < NEXP; ++e) sum += __expf(lrow[e] - m);
        const float invSum = 1.0f / sum;

        unsigned long long sel = 0ull;
        float hsNum = 0.0f;
        for (int it = 0; it < TOPK; ++it) {
            float best = -INFINITY; int bi = 0;
            for (int e = 0; e < NEXP; ++e) {
                if (!((sel >> e) & 1ull)) {
                    float x = lrow[e];
                    if (x > best) { best = x; bi = e; }
                }
            }
            sel   |= (1ull << bi);
            hsNum += __expf(best - m);
        }
        const float rinv = 1.0f / (hsNum * invSum + 1e-9f);

        float* oh = out_hard + (size_t)grow * NEXP;
        float* os = out_soft + (size_t)grow * NEXP;
        #pragma unroll 4
        for (int e = 0; e < NEXP; e += 4) {
            float s0 = __expf(lrow[e + 0] - m) * invSum;
            float s1 = __expf(lrow[e + 1] - m) * invSum;
            float s2 = __expf(lrow[e + 2] - m) * invSum;
            float s3 = __expf(lrow[e + 3] - m) * invSum;
            float4 s4 = make_float4(s0, s1, s2, s3);
            float4 h4 = make_float4(((sel >> (e + 0)) & 1ull) ? s0 * rinv : 0.0f,
                                    ((sel >> (e + 1)) & 1ull) ? s1 * rinv : 0.0f,
                                    ((sel >> (e + 2)) & 1ull) ? s2 * rinv : 0.0f,
                                    ((sel >> (e + 3)) & 1ull) ? s3 * rinv : 0.0f);
            *(float4*)(os + e) = s4;
            *(float4*)(oh + e) = h4;
        }
    }
}

extern "C" void kernel_launch(void* const* d_in, const int* in_sizes, int n_in,
                              void* d_out, int out_size, void* d_ws, size_t ws_size,
                              hipStream_t stream) {
    const float* r    = (const float*)d_in[0];   // (32768, 2048)
    const float* W    = (const float*)d_in[1];   // (64, 2048)
    const float* bias = (const float*)d_in[2];   // (64,)
    float* out_hard = (float*)d_out;                      // first tuple element
    float* out_soft = out_hard + (size_t)B_ROWS * NEXP;   // second tuple element

    const size_t smemBytes = 2 * (size_t)BUF_BYTES;       // 133120 B
    dim3 grid(B_ROWS / ROWS_PER_BLK);   // 256
    dim3 block(THREADS);                // 256 threads = 8 waves (wave32)
    moe_gate_wmma_kernel<<<grid, block, smemBytes, stream>>>(r, W, bias,
                                                             out_hard, out_soft);
}